// SeqSelfAttention_29626684407936
// MI455X (gfx1250) — compile-verified
//
#include <hip/hip_runtime.h>
#include <math.h>

typedef float v2f __attribute__((ext_vector_type(2)));
typedef float v8f __attribute__((ext_vector_type(8)));
typedef int   v4i __attribute__((ext_vector_type(4)));

constexpr int Bc = 4;
constexpr int Lc = 1024;
constexpr int Dc = 512;
constexpr int Hc = 64;
constexpr int QTILE = 16;
constexpr int KCHUNK = 128;
constexpr int KSTR = Hc + 4;      // 68 floats: 16B-aligned rows + bank-skewed
constexpr int TPB = 512;          // 16 waves (wave32)
constexpr int NCHUNK = Lc / KCHUNK;

// ---------- fast math on CDNA5 ----------
__device__ __forceinline__ float fast_tanh(float x) {
#if __has_builtin(__builtin_amdgcn_tanhf)
  return __builtin_amdgcn_tanhf(x);      // v_tanh_f32 (TRANS pipe)
#else
  return tanhf(x);
#endif
}
__device__ __forceinline__ float fast_exp(float x) {
#if __has_builtin(__builtin_amdgcn_exp2f)
  return __builtin_amdgcn_exp2f(x * 1.4426950408889634f);  // v_exp_f32
#else
  return expf(x);
#endif
}
__device__ __forceinline__ float fast_rcp(float x) {
#if __has_builtin(__builtin_amdgcn_rcpf)
  return __builtin_amdgcn_rcpf(x);       // v_rcp_f32
#else
  return 1.0f / x;
#endif
}

// ---------- async global->LDS (CDNA5 GLOBAL_LOAD_ASYNC_TO_LDS_B128) ----------
#if __has_builtin(__builtin_amdgcn_global_load_async_to_lds_b128)
#define HAS_ASYNC_LDS 1
#endif

__device__ __forceinline__ void cp16_g2l(const float* __restrict__ g, float* l) {
#ifdef HAS_ASYNC_LDS
  __builtin_amdgcn_global_load_async_to_lds_b128(
      (__attribute__((address_space(1))) v4i*)g,
      (__attribute__((address_space(3))) v4i*)l, /*offset=*/0, /*cpol=*/0);
#else
  *(float4*)l = *(const float4*)g;
#endif
}
__device__ __forceinline__ void wait_async_all() {
#ifdef HAS_ASYNC_LDS
#if __has_builtin(__builtin_amdgcn_s_wait_asynccnt)
  __builtin_amdgcn_s_wait_asynccnt(0);
#else
  asm volatile("s_wait_asynccnt 0" ::: "memory");
#endif
#endif
}

// =====================================================================
// Kernel 1: Q = x*Wt + bh ; K = x*Wx   via V_WMMA_F32_16X16X4_F32
// One wave per 16-row tile of the (B*L, D) activation matrix.
// =====================================================================
__global__ __launch_bounds__(32)
void proj_qk_kernel(const float* __restrict__ x,
                    const float* __restrict__ Wt,
                    const float* __restrict__ Wx,
                    const float* __restrict__ bh,
                    float* __restrict__ Qws,
                    float* __restrict__ Kws) {
  const int row0  = blockIdx.x * QTILE;        // 256 tiles over B*L = 4096 rows
  const int lane  = threadIdx.x;               // 0..31
  const int m     = lane & 15;                 // row within tile / col within N-tile
  const int khalf = (lane >> 4) * 2;           // 0 or 2: K sub-offset per lane half

  v8f accQ[4] = {{0.f}}, accK[4] = {{0.f}};    // 4 N-tiles of 16 cols (H = 64)

  for (int d = 0; d < Dc; d += 4) {
    // A fragment: lane holds (M=m, K=d+khalf .. d+khalf+1)
    const float* xr = x + (size_t)(row0 + m) * Dc + d + khalf;
    v2f a; a.x = xr[0]; a.y = xr[1];
#pragma unroll
    for (int nt = 0; nt < 4; ++nt) {
      const int col = nt * 16 + m;
      const float* wtp = Wt + (size_t)(d + khalf) * Hc + col;
      const float* wxp = Wx + (size_t)(d + khalf) * Hc + col;
      v2f bt; bt.x = wtp[0]; bt.y = wtp[Hc];
      v2f bx; bx.x = wxp[0]; bx.y = wxp[Hc];
      accQ[nt] = __builtin_amdgcn_wmma_f32_16x16x4_f32(
          false, a, false, bt, (short)0, accQ[nt], false, false);
      accK[nt] = __builtin_amdgcn_wmma_f32_16x16x4_f32(
          false, a, false, bx, (short)0, accK[nt], false, false);
    }
  }

  // C/D layout: vgpr i -> row i (lanes 0-15) / row i+8 (lanes 16-31), col = nt*16+m
#pragma unroll
  for (int nt = 0; nt < 4; ++nt) {
    const int col  = nt * 16 + m;
    const float bias = bh[col];                 // fold bh into Q (tanh(q+k+bh))
#pragma unroll
    for (int i = 0; i < 8; ++i) {
      const int r = (lane < 16) ? i : (i + 8);
      Qws[(size_t)(row0 + r) * Hc + col] = accQ[nt][i] + bias;
      Kws[(size_t)(row0 + r) * Hc + col] = accK[nt][i];
    }
  }
}

// =====================================================================
// Kernel 2: fused  e -> sigmoid -> softmax -> context(a @ x)
// One 512-thread block (16 waves) per (batch, 16-query tile).
// K chunks double-buffered in LDS via async DMA, overlapped with tanh.
// =====================================================================
__global__ __launch_bounds__(TPB)
void attn_kernel(const float* __restrict__ x,
                 const float* __restrict__ Qws,
                 const float* __restrict__ Kws,
                 const float* __restrict__ Wa,
                 const float* __restrict__ ba,
                 float* __restrict__ out) {
  __shared__ alignas(16) float Qs[QTILE * (Hc + 1)];    //  4.2 KB
  __shared__ alignas(16) float KsD[2][KCHUNK * KSTR];   // 69.6 KB (double buffer)
  __shared__ alignas(16) float Sb[QTILE * (Lc + 1)];    // 65.6 KB score rows
  __shared__ float WaS[Hc];
  __shared__ float rinv[QTILE];

  const int b   = blockIdx.x >> 6;              // 64 q-tiles per batch
  const int q0  = (blockIdx.x & 63) * QTILE;
  const int tid = threadIdx.x;
  const int wave = tid >> 5, lane = tid & 31;
  const float bav = ba[0];

  const float* kgbase = Kws + (size_t)b * Lc * Hc;

  // stage one 128x64 K chunk (2048 x float4, 4 per thread) into an LDS buffer
  auto stage_chunk = [&](int c, float* buf) {
    const float* src = kgbase + (size_t)c * KCHUNK * Hc;
#pragma unroll
    for (int j = 0; j < (KCHUNK * Hc / 4) / TPB; ++j) {   // 4 iterations
      const int i   = tid + j * TPB;                      // float4 index 0..2047
      const int row = i >> 4;
      const int h4  = (i & 15) * 4;
      cp16_g2l(src + (size_t)row * Hc + h4, &buf[row * KSTR + h4]);
    }
  };

  // stage Q tile (+bh already folded) and Wa
  for (int i = tid; i < QTILE * Hc; i += TPB) {
    const int q = i >> 6, h = i & 63;
    Qs[q * (Hc + 1) + h] = Qws[(size_t)(b * Lc + q0 + q) * Hc + h];
  }
  if (tid < Hc) WaS[tid] = Wa[tid];

  // ---- e / sigmoid phase over 128-key chunks, double buffered ----
  const int q     = tid & 15;                   // query owned by this thread
  const int kbase = tid >> 4;                   // 0..31; keys kbase + p*32

  stage_chunk(0, KsD[0]);                       // prologue DMA
  for (int c = 0; c < NCHUNK; ++c) {
    wait_async_all();                           // chunk c landed (this wave's DMAs)
    __syncthreads();                            // all waves: chunk c visible,
                                                // chunk c-1 compute finished
    if (c + 1 < NCHUNK) stage_chunk(c + 1, KsD[(c + 1) & 1]);  // overlap DMA

    const float* kb = KsD[c & 1];
    float e0 = bav, e1 = bav, e2 = bav, e3 = bav;
    const float* qrow = &Qs[q * (Hc + 1)];
    const float* kr0 = kb + (kbase +  0) * KSTR;
    const float* kr1 = kb + (kbase + 32) * KSTR;
    const float* kr2 = kb + (kbase + 64) * KSTR;
    const float* kr3 = kb + (kbase + 96) * KSTR;
#pragma unroll 8
    for (int h = 0; h < Hc; ++h) {
      const float w = WaS[h], qv = qrow[h];     // 4 independent v_tanh chains
      e0 += w * fast_tanh(qv + kr0[h]);
      e1 += w * fast_tanh(qv + kr1[h]);
      e2 += w * fast_tanh(qv + kr2[h]);
      e3 += w * fast_tanh(qv + kr3[h]);
    }
    float* srow = &Sb[q * (Lc + 1) + c * KCHUNK + kbase];
    srow[ 0] = fast_rcp(1.0f + fast_exp(-e0));  // sigmoid
    srow[32] = fast_rcp(1.0f + fast_exp(-e1));
    srow[64] = fast_rcp(1.0f + fast_exp(-e2));
    srow[96] = fast_rcp(1.0f + fast_exp(-e3));
  }
  __syncthreads();

  // ---- softmax over keys: one wave per query row ----
  {
    float sum = 0.f;
    float* srow = &Sb[wave * (Lc + 1)];
    for (int k = lane; k < Lc; k += 32) {
      const float ex = fast_exp(srow[k]);
      srow[k] = ex;
      sum += ex;
    }
#pragma unroll
    for (int off = 16; off > 0; off >>= 1) sum += __shfl_xor(sum, off, 32);
    if (lane == 0) rinv[wave] = fast_rcp(sum);
  }
  __syncthreads();
  for (int i = tid; i < QTILE * Lc; i += TPB) {
    const int qq = i >> 10, k = i & 1023;
    Sb[qq * (Lc + 1) + k] *= rinv[qq];
  }
  __syncthreads();

  // ---- context GEMM: C[16,512] = A[16,1024] (LDS) x x_b[1024,512] ----
  const int m     = lane & 15;
  const int kr    = (lane >> 4) * 2;
  const float* xb = x + (size_t)b * Lc * Dc;
#pragma unroll
  for (int t = 0; t < 2; ++t) {                 // 16 waves x 2 = 32 N-tiles
    const int ncol = (wave * 2 + t) * 16 + m;
    v8f acc = {0.f};
    for (int k = 0; k < Lc; k += 4) {
      v2f a;  a.x = Sb[m * (Lc + 1) + k + kr];
              a.y = Sb[m * (Lc + 1) + k + kr + 1];
      v2f bb; bb.x = xb[(size_t)(k + kr)     * Dc + ncol];
              bb.y = xb[(size_t)(k + kr + 1) * Dc + ncol];
      acc = __builtin_amdgcn_wmma_f32_16x16x4_f32(
          false, a, false, bb, (short)0, acc, false, false);
    }
#pragma unroll
    for (int i = 0; i < 8; ++i) {
      const int r = (lane < 16) ? i : (i + 8);
      out[(size_t)(b * Lc + q0 + r) * Dc + ncol] = acc[i];
    }
  }
}

// =====================================================================
extern "C" void kernel_launch(void* const* d_in, const int* in_sizes, int n_in,
                              void* d_out, int out_size, void* d_ws, size_t ws_size,
                              hipStream_t stream) {
  const float* x  = (const float*)d_in[0];
  const float* Wt = (const float*)d_in[1];
  const float* Wx = (const float*)d_in[2];
  const float* bh = (const float*)d_in[3];
  const float* Wa = (const float*)d_in[4];
  const float* ba = (const float*)d_in[5];
  float* out = (float*)d_out;

  float* Qws = (float*)d_ws;                          // B*L*H floats
  float* Kws = Qws + (size_t)Bc * Lc * Hc;            // B*L*H floats (2 MB total)

  proj_qk_kernel<<<(Bc * Lc) / QTILE, 32, 0, stream>>>(x, Wt, Wx, bh, Qws, Kws);
  attn_kernel<<<Bc * (Lc / QTILE), TPB, 0, stream>>>(x, Qws, Kws, Wa, ba, out);
}